// GNNmodel_85770496901295
// MI455X (gfx1250) — compile-verified
//
#include <hip/hip_runtime.h>
#include <hip/hip_bf16.h>

typedef float v2f __attribute__((ext_vector_type(2)));
typedef float v8f __attribute__((ext_vector_type(8)));

#define THREADS 256

// ---------------------------------------------------------------------------
// Zero-fill (graph-capture-safe replacement for hipMemsetAsync)
// ---------------------------------------------------------------------------
__global__ __launch_bounds__(THREADS)
void zero_kernel(float* __restrict__ p, int n) {
    int i = blockIdx.x * THREADS + threadIdx.x;
    if (i < n) p[i] = 0.0f;
}

// ---------------------------------------------------------------------------
// Small-matrix transpose: W[F][K] -> WT[K][F]
// ---------------------------------------------------------------------------
__global__ __launch_bounds__(THREADS)
void transpose_kernel(const float* __restrict__ W, float* __restrict__ WT,
                      int F, int K) {
    int i = blockIdx.x * THREADS + threadIdx.x;
    if (i >= F * K) return;
    int f = i / K;
    int k = i - f * K;
    WT[(size_t)k * F + f] = W[i];
}

// ---------------------------------------------------------------------------
// Edge scatter-add: OUT[dst[e]][:] += X[src[e]][:]  (d multiple of 4)
// one thread per (edge, float4 chunk); native fp32 atomics (non-returning)
// ---------------------------------------------------------------------------
__global__ __launch_bounds__(THREADS)
void scatter_add_kernel(const float* __restrict__ X,
                        const int* __restrict__ src,
                        const int* __restrict__ dst,
                        float* __restrict__ OUT,
                        int nE, int d4) {
    long long t = (long long)blockIdx.x * THREADS + threadIdx.x;
    long long total = (long long)nE * d4;
    if (t >= total) return;
    int e = (int)(t / d4);
    int c = (int)(t - (long long)e * d4);
    int d = d4 * 4;
    int s = src[e];
    int q = dst[e];
    const float4 v = *(const float4*)(X + (size_t)s * d + (size_t)c * 4);
    float* o = OUT + (size_t)q * d + (size_t)c * 4;
    unsafeAtomicAdd(o + 0, v.x);
    unsafeAtomicAdd(o + 1, v.y);
    unsafeAtomicAdd(o + 2, v.z);
    unsafeAtomicAdd(o + 3, v.w);
}

// ---------------------------------------------------------------------------
// Graph pooling: G[batch[i]][:] += H[i][:]   (d multiple of 4)
// ---------------------------------------------------------------------------
__global__ __launch_bounds__(THREADS)
void pool_kernel(const float* __restrict__ H,
                 const int* __restrict__ batch,
                 float* __restrict__ G,
                 int n, int d4) {
    long long t = (long long)blockIdx.x * THREADS + threadIdx.x;
    long long total = (long long)n * d4;
    if (t >= total) return;
    int i = (int)(t / d4);
    int c = (int)(t - (long long)i * d4);
    int d = d4 * 4;
    int g = batch[i];
    const float4 v = *(const float4*)(H + (size_t)i * d + (size_t)c * 4);
    float* o = G + (size_t)g * d + (size_t)c * 4;
    unsafeAtomicAdd(o + 0, v.x);
    unsafeAtomicAdd(o + 1, v.y);
    unsafeAtomicAdd(o + 2, v.z);
    unsafeAtomicAdd(o + 3, v.w);
}

// ---------------------------------------------------------------------------
// Fused GraphConv node update via fp32 WMMA:
//   OUT = A @ WrelT (+ X @ WrootT) + bias, optional ReLU
//
// Templated on K/F so the K-loop fully unrolls; each wave preloads its entire
// B operand (K x 16 slice of WrelT / WrootT) into registers, then the hot loop
// is pure ds_load + v_wmma. Waves cover (F/16) n-tiles x (8/(F/16)) m-subtiles
// so all 8 waves compute for every layer shape. Grid-stride over 16*MW-row
// m-groups amortizes the B preload.
//
// V_WMMA_F32_16X16X4_F32 fragment layout (ISA 7.12.2):
//   A 16x4 :  lane&15 = M ; vgpr j holds K = j + 2*(lane>=16)
//   B 4x16 :  lane&15 = N ; vgpr j holds K = j + 2*(lane>=16)
//   C/D    :  lane&15 = N ; vgpr r holds M = r + 8*(lane>=16)
// ---------------------------------------------------------------------------
template<int K, int F, bool HAS_ROOT, bool RELU>
__global__ __launch_bounds__(THREADS)
void gemm_conv_kernel(const float* __restrict__ A,
                      const float* __restrict__ X,       // unused if !HAS_ROOT
                      const float* __restrict__ WrelT,   // [K,F]
                      const float* __restrict__ WrootT,  // [K,F], unused if !HAS_ROOT
                      const float* __restrict__ bias,    // [F]
                      float* __restrict__ OUT,           // [M,F]
                      int M) {
    constexpr int NT = F / 16;        // n-tiles across waves
    constexpr int MW = 8 / NT;        // m-subtiles per block iteration
    constexpr int R  = 16 * MW;       // rows staged per iteration
    constexpr int KS = K / 4;         // wmma K-steps

    __shared__ float ldsA[R][K + 1];                     // +1 pad: no bank conflicts
    __shared__ float ldsX[HAS_ROOT ? R : 1][K + 1];

    const int tid  = threadIdx.x;
    const int wave = tid >> 5;
    const int lane = tid & 31;
    const int half = lane >> 4;       // 0: lanes 0-15, 1: lanes 16-31
    const int l15  = lane & 15;
    const int n0   = (wave % NT) * 16;
    const int msub = wave / NT;       // which 16-row subtile of the group

    // ---- preload this wave's B fragments into registers (stays hot in L2) --
    v2f brel[KS];
#pragma unroll
    for (int s = 0; s < KS; ++s) {
        const int k = s * 4 + half * 2;
        brel[s][0] = WrelT[(size_t)(k + 0) * F + n0 + l15];
        brel[s][1] = WrelT[(size_t)(k + 1) * F + n0 + l15];
    }
    v2f broot[HAS_ROOT ? KS : 1];
    if constexpr (HAS_ROOT) {
#pragma unroll
        for (int s = 0; s < KS; ++s) {
            const int k = s * 4 + half * 2;
            broot[s][0] = WrootT[(size_t)(k + 0) * F + n0 + l15];
            broot[s][1] = WrootT[(size_t)(k + 1) * F + n0 + l15];
        }
    }
    const float bv = bias[n0 + l15];

    const int nGroups = (M + R - 1) / R;
    for (int grp = blockIdx.x; grp < nGroups; grp += gridDim.x) {
        const int row0 = grp * R;

        // ---- stage A (and X) rows [row0, row0+R) into LDS, coalesced ----
        for (int i = tid; i < R * K; i += THREADS) {
            const int r = i / K, c = i - r * K;
            const int gr = row0 + r;
            const bool ok = (gr < M);
            ldsA[r][c] = ok ? A[(size_t)gr * K + c] : 0.0f;
            if constexpr (HAS_ROOT)
                ldsX[r][c] = ok ? X[(size_t)gr * K + c] : 0.0f;
        }
        // prefetch next group's rows while we compute this one
        {
            const int nrow0 = row0 + (int)gridDim.x * R;
            if (nrow0 < M) {
                for (int r = tid; r < R; r += THREADS) {
                    if (nrow0 + r < M) {
                        __builtin_prefetch(&A[(size_t)(nrow0 + r) * K], 0, 3);
                        if constexpr (HAS_ROOT)
                            __builtin_prefetch(&X[(size_t)(nrow0 + r) * K], 0, 3);
                    }
                }
            }
        }
        __syncthreads();

        v8f acc;
#pragma unroll
        for (int r = 0; r < 8; ++r) acc[r] = bv;

        const int arow = msub * 16 + l15;
#pragma unroll
        for (int s = 0; s < KS; ++s) {
            const int k = s * 4 + half * 2;
            v2f a;
            a[0] = ldsA[arow][k + 0];
            a[1] = ldsA[arow][k + 1];
            acc = __builtin_amdgcn_wmma_f32_16x16x4_f32(false, a, false, brel[s],
                                                        (short)0, acc, false, false);
            if constexpr (HAS_ROOT) {
                v2f ax;
                ax[0] = ldsX[arow][k + 0];
                ax[1] = ldsX[arow][k + 1];
                acc = __builtin_amdgcn_wmma_f32_16x16x4_f32(false, ax, false, broot[s],
                                                            (short)0, acc, false, false);
            }
        }

        const int mbase = row0 + msub * 16 + half * 8;
#pragma unroll
        for (int r = 0; r < 8; ++r) {
            float v = acc[r];
            if constexpr (RELU) v = fmaxf(v, 0.0f);
            if (mbase + r < M)
                OUT[(size_t)(mbase + r) * F + n0 + l15] = v;
        }
        __syncthreads();
    }
}

// ---------------------------------------------------------------------------
// Row-wise log-softmax, 32 classes = exactly one wave32 per row
// ---------------------------------------------------------------------------
__global__ __launch_bounds__(THREADS)
void logsoftmax_kernel(const float* __restrict__ S,
                       float* __restrict__ OUT, int nrows) {
    const int wave = threadIdx.x >> 5;
    const int lane = threadIdx.x & 31;
    const int row  = blockIdx.x * 8 + wave;
    if (row >= nrows) return;
    float v = S[(size_t)row * 32 + lane];
    float m = v;
#pragma unroll
    for (int off = 16; off > 0; off >>= 1)
        m = fmaxf(m, __shfl_xor(m, off, 32));
    float e = expf(v - m);
    float s = e;
#pragma unroll
    for (int off = 16; off > 0; off >>= 1)
        s += __shfl_xor(s, off, 32);
    OUT[(size_t)row * 32 + lane] = v - m - logf(s);
}

// ---------------------------------------------------------------------------
// Host-side orchestration
// ---------------------------------------------------------------------------
static inline int ceil_div(long long a, int b) { return (int)((a + b - 1) / b); }
static inline int min_i(int a, int b) { return a < b ? a : b; }

extern "C" void kernel_launch(void* const* d_in, const int* in_sizes, int n_in,
                              void* d_out, int out_size, void* d_ws, size_t ws_size,
                              hipStream_t stream) {
    const float* x       = (const float*)d_in[0];
    const int*   edge    = (const int*)  d_in[1];
    const int*   batch   = (const int*)  d_in[2];
    const float* w1_rel  = (const float*)d_in[3];
    const float* b1      = (const float*)d_in[4];
    const float* w1_root = (const float*)d_in[5];
    const float* w2_rel  = (const float*)d_in[6];
    const float* b2      = (const float*)d_in[7];
    const float* w2_root = (const float*)d_in[8];
    const float* w3_rel  = (const float*)d_in[9];
    const float* b3      = (const float*)d_in[10];
    const float* w3_root = (const float*)d_in[11];
    const float* fc1_w   = (const float*)d_in[12];
    const float* fc1_b   = (const float*)d_in[13];
    const float* fc2_w   = (const float*)d_in[14];
    const float* fc2_b   = (const float*)d_in[15];

    const int N  = in_sizes[2];          // 50000 nodes
    const int nE = in_sizes[1] / 2;      // 600000 edges
    const int G  = 512;
    const int* src = edge;
    const int* dst = edge + nE;

    float* ws = (float*)d_ws;
    size_t off = 0;
    auto alloc = [&](size_t n) { float* p = ws + off; off += n; return p; };
    float* aggr    = alloc((size_t)N * 128);
    float* h1      = alloc((size_t)N * 128);
    float* h2      = alloc((size_t)N * 64);
    float* h3      = alloc((size_t)N * 64);
    float* g       = alloc((size_t)G * 64);
    float* t       = alloc((size_t)G * 64);
    float* sc      = alloc((size_t)G * 32);
    float* w1relT  = alloc(128 * 128);
    float* w1rootT = alloc(128 * 128);
    float* w2relT  = alloc(128 * 64);
    float* w2rootT = alloc(128 * 64);
    float* w3relT  = alloc(64 * 64);
    float* w3rootT = alloc(64 * 64);
    float* fc1T    = alloc(64 * 64);
    float* fc2T    = alloc(64 * 32);

    // ---- weight transposes (tiny) ----
    transpose_kernel<<<ceil_div(128 * 128, THREADS), THREADS, 0, stream>>>(w1_rel,  w1relT,  128, 128);
    transpose_kernel<<<ceil_div(128 * 128, THREADS), THREADS, 0, stream>>>(w1_root, w1rootT, 128, 128);
    transpose_kernel<<<ceil_div(64 * 128,  THREADS), THREADS, 0, stream>>>(w2_rel,  w2relT,  64, 128);
    transpose_kernel<<<ceil_div(64 * 128,  THREADS), THREADS, 0, stream>>>(w2_root, w2rootT, 64, 128);
    transpose_kernel<<<ceil_div(64 * 64,   THREADS), THREADS, 0, stream>>>(w3_rel,  w3relT,  64, 64);
    transpose_kernel<<<ceil_div(64 * 64,   THREADS), THREADS, 0, stream>>>(w3_root, w3rootT, 64, 64);
    transpose_kernel<<<ceil_div(64 * 64,   THREADS), THREADS, 0, stream>>>(fc1_w,   fc1T,    64, 64);
    transpose_kernel<<<ceil_div(32 * 64,   THREADS), THREADS, 0, stream>>>(fc2_w,   fc2T,    32, 64);

    const int MAXBLK = 1024;

    // ---- layer 1: K=128, F=128, relu ----
    zero_kernel<<<ceil_div((long long)N * 128, THREADS), THREADS, 0, stream>>>(aggr, N * 128);
    scatter_add_kernel<<<ceil_div((long long)nE * 32, THREADS), THREADS, 0, stream>>>(x, src, dst, aggr, nE, 32);
    gemm_conv_kernel<128, 128, true, true>
        <<<min_i(ceil_div(N, 16), MAXBLK), THREADS, 0, stream>>>(aggr, x, w1relT, w1rootT, b1, h1, N);

    // ---- layer 2: K=128, F=64, relu ----
    zero_kernel<<<ceil_div((long long)N * 128, THREADS), THREADS, 0, stream>>>(aggr, N * 128);
    scatter_add_kernel<<<ceil_div((long long)nE * 32, THREADS), THREADS, 0, stream>>>(h1, src, dst, aggr, nE, 32);
    gemm_conv_kernel<128, 64, true, true>
        <<<min_i(ceil_div(N, 32), MAXBLK), THREADS, 0, stream>>>(aggr, h1, w2relT, w2rootT, b2, h2, N);

    // ---- layer 3: K=64, F=64, no relu ----
    zero_kernel<<<ceil_div((long long)N * 64, THREADS), THREADS, 0, stream>>>(aggr, N * 64);
    scatter_add_kernel<<<ceil_div((long long)nE * 16, THREADS), THREADS, 0, stream>>>(h2, src, dst, aggr, nE, 16);
    gemm_conv_kernel<64, 64, true, false>
        <<<min_i(ceil_div(N, 32), MAXBLK), THREADS, 0, stream>>>(aggr, h2, w3relT, w3rootT, b3, h3, N);

    // ---- global add pool ----
    zero_kernel<<<ceil_div((long long)G * 64, THREADS), THREADS, 0, stream>>>(g, G * 64);
    pool_kernel<<<ceil_div((long long)N * 16, THREADS), THREADS, 0, stream>>>(h3, batch, g, N, 16);

    // ---- fc1 (relu) / fc2 ----
    gemm_conv_kernel<64, 64, false, true>
        <<<ceil_div(G, 32), THREADS, 0, stream>>>(g, nullptr, fc1T, nullptr, fc1_b, t, G);
    gemm_conv_kernel<64, 32, false, false>
        <<<ceil_div(G, 64), THREADS, 0, stream>>>(t, nullptr, fc2T, nullptr, fc2_b, sc, G);

    // ---- log-softmax ----
    logsoftmax_kernel<<<ceil_div(G, 8), THREADS, 0, stream>>>(sc, (float*)d_out, G);
}